// Loss_generator_28570122453293
// MI455X (gfx1250) — compile-verified
//
#include <hip/hip_runtime.h>
#include <stdint.h>

// ---------------- problem constants (from reference) ----------------
#define NT    198            // T - 2 residual time steps
#define Hh    512
#define Ww    512
#define DOM   513            // wrap-padded output domain per spatial dim
#define DT_F  0.0125f
#define DX_F  0.2f
#define MU_F  0.1f
#define BETA_F 1.0f

// ---------------- tiling ----------------
#define TSTEPS  8                          // timesteps chained per block
#define NTCH    ((NT + TSTEPS - 1) / TSTEPS)   // 25 chunks
#define TH      16                         // output rows per tile
#define TW      128                        // output cols per tile
#define PR      (TH + 4)                   // 20 (2-wide halo)
#define PC      (TW + 4)                   // 132
#define TILE_N  (PR * PC)                  // 2640 elems per channel
#define NYB     ((DOM + TH - 1) / TH)      // 33
#define NXB     ((DOM + TW - 1) / TW)      // 5
#define NBLOCKS (NTCH * NYB * NXB)         // 4125
#define THREADS 256
#define FRAME   ((size_t)2 * Hh * Ww)

// ---------------- CDNA5 async global->LDS copy path ----------------
#if defined(__gfx1250__) && __has_builtin(__builtin_amdgcn_global_load_async_to_lds_b32)
typedef __attribute__((address_space(1))) int g_as_int;   // global
typedef __attribute__((address_space(3))) int l_as_int;   // LDS

__device__ __forceinline__ void async_cp_f32(const float* gsrc, float* ldst) {
    __builtin_amdgcn_global_load_async_to_lds_b32(
        (g_as_int*)(uintptr_t)gsrc,
        (l_as_int*)(uint32_t)(uintptr_t)ldst,
        0, 0);
}
__device__ __forceinline__ void async_wait_all() {
#if __has_builtin(__builtin_amdgcn_s_wait_asynccnt)
    __builtin_amdgcn_s_wait_asynccnt(0);
#else
    asm volatile("s_wait_asynccnt 0" ::: "memory");
#endif
}
#else
__device__ __forceinline__ void async_cp_f32(const float* gsrc, float* ldst) {
    *ldst = *gsrc;           // fallback: VGPR round-trip copy
}
__device__ __forceinline__ void async_wait_all() {}
#endif

// Issue async loads of one haloed (PR x PC) tile for both channels of frame f.
__device__ __forceinline__ void issue_tile(const float* __restrict__ src, int f,
                                           int y0, int x0,
                                           float* dstU, float* dstV, int tid) {
    const float* ub = src + (size_t)f * FRAME;       // channel 0
    const float* vb = ub + (size_t)Hh * Ww;          // channel 1
    for (int e = tid; e < TILE_N; e += THREADS) {
        const int r  = e / PC;
        const int c  = e - r * PC;
        const int gr = (y0 + r - 2 + Hh) & (Hh - 1); // periodic wrap
        const int gc = (x0 + c - 2 + Ww) & (Ww - 1);
        const int gi = gr * Ww + gc;
        async_cp_f32(ub + gi, dstU + e);
        async_cp_f32(vb + gi, dstV + e);
    }
}

// ---- stage 1: chained stencil + residual + per-block partial MSE ----
__global__ __launch_bounds__(THREADS)
void gs_residual_partial(const float* __restrict__ src, float* __restrict__ psum) {
    const int t0   = blockIdx.x * TSTEPS;
    const int tend = (t0 + TSTEPS < NT) ? (t0 + TSTEPS) : NT;
    const int y0   = blockIdx.y * TH;
    const int x0   = blockIdx.z * TW;
    const int tid  = threadIdx.x;

    __shared__ float buf[3][2][TILE_N];              // 63,360 B
    __shared__ float red[2 * (THREADS / 32)];

    // prime the pipeline: frames t0 and t0+1
    issue_tile(src, t0,     y0, x0, buf[0][0], buf[0][1], tid);
    issue_tile(src, t0 + 1, y0, x0, buf[1][0], buf[1][1], tid);
    async_wait_all();
    __syncthreads();

    const float inv_dx2 = 1.0f / (DX_F * DX_F);
    const float c0 = -5.0f * inv_dx2;
    const float c1 = (4.0f / 3.0f) * inv_dx2;
    const float c2 = (-1.0f / 12.0f) * inv_dx2;
    const float inv_dt = 1.0f / DT_F;

    float su = 0.0f, sv = 0.0f;

    for (int k = 0; t0 + k < tend; ++k) {
        const int t   = t0 + k;
        const int cur = k % 3;
        const int nxt = (k + 1) % 3;
        const int pre = (k + 2) % 3;

        // overlap: DMA frame t+2 into the buffer freed by the last barrier
        if (t + 1 < tend)
            issue_tile(src, t + 2, y0, x0, buf[pre][0], buf[pre][1], tid);

        const float* Ut = buf[cur][0];
        const float* Vt = buf[cur][1];
        const float* Un = buf[nxt][0];
        const float* Vn = buf[nxt][1];

        for (int e = tid; e < TH * TW; e += THREADS) {
            const int r = e >> 7;                    // e / TW
            const int c = e & (TW - 1);              // e % TW
            const int i = y0 + r;
            const int j = x0 + c;
            if (i >= DOM || j >= DOM) continue;

            const int lr = r + 2, lc = c + 2;
            const int idx = lr * PC + lc;
            const float uc  = Ut[idx];
            const float vc  = Vt[idx];
            const float un1 = Ut[idx - PC] + Ut[idx + PC] + Ut[idx - 1] + Ut[idx + 1];
            const float un2 = Ut[idx - 2 * PC] + Ut[idx + 2 * PC] + Ut[idx - 2] + Ut[idx + 2];
            const float vn1 = Vt[idx - PC] + Vt[idx + PC] + Vt[idx - 1] + Vt[idx + 1];
            const float vn2 = Vt[idx - 2 * PC] + Vt[idx + 2 * PC] + Vt[idx - 2] + Vt[idx + 2];
            const float lap_u = c0 * uc + c1 * un1 + c2 * un2;
            const float lap_v = c0 * vc + c1 * vn1 + c2 * vn2;

            const float ut = (Un[idx] - uc) * inv_dt; // t+1 center from LDS
            const float vt = (Vn[idx] - vc) * inv_dt;

            const float s  = uc * uc + vc * vc;
            const float fu = ut - (MU_F * lap_u + (1.0f - s) * uc + BETA_F * s * vc);
            const float fv = vt - (MU_F * lap_v + (1.0f - s) * vc - BETA_F * s * uc);
            su += fu * fu;
            sv += fv * fv;
        }

        async_wait_all();       // frame t+2 landed
        __syncthreads();        // everyone done reading cur before it is recycled
    }

    // ---- wave32 reduce, then cross-wave via LDS ----
    for (int off = 16; off > 0; off >>= 1) {
        su += __shfl_down(su, off, 32);
        sv += __shfl_down(sv, off, 32);
    }
    const int lane = tid & 31, wid = tid >> 5;
    if (lane == 0) { red[wid] = su; red[(THREADS / 32) + wid] = sv; }
    __syncthreads();
    if (tid == 0) {
        float tu = 0.0f, tv = 0.0f;
        for (int w = 0; w < THREADS / 32; ++w) {
            tu += red[w];
            tv += red[(THREADS / 32) + w];
        }
        const int b = ((int)blockIdx.x * NYB + (int)blockIdx.y) * NXB + (int)blockIdx.z;
        psum[b]           = tu;
        psum[NBLOCKS + b] = tv;
    }
}

// ---------------- stage 2: deterministic f64 reduce -> mean ----------------
__global__ __launch_bounds__(THREADS)
void gs_reduce_mean(const float* __restrict__ psum, float* __restrict__ out) {
    const int ch = blockIdx.x;                 // 0 = u, 1 = v
    const float* p = psum + (size_t)ch * NBLOCKS;

    double acc = 0.0;
    for (int i = threadIdx.x; i < NBLOCKS; i += THREADS)
        acc += (double)p[i];

    for (int off = 16; off > 0; off >>= 1)
        acc += __shfl_down(acc, off, 32);

    __shared__ double sred[THREADS / 32];
    const int lane = threadIdx.x & 31, wid = threadIdx.x >> 5;
    if (lane == 0) sred[wid] = acc;
    __syncthreads();
    if (threadIdx.x == 0) {
        double tot = 0.0;
        for (int w = 0; w < THREADS / 32; ++w) tot += sred[w];
        const double N = (double)NT * (double)DOM * (double)DOM;
        out[ch] = (float)(tot / N);
    }
}

extern "C" void kernel_launch(void* const* d_in, const int* in_sizes, int n_in,
                              void* d_out, int out_size, void* d_ws, size_t ws_size,
                              hipStream_t stream) {
    (void)in_sizes; (void)n_in; (void)out_size; (void)ws_size;
    const float* src = (const float*)d_in[0];
    float* out = (float*)d_out;
    float* ws  = (float*)d_ws;       // needs 2*NBLOCKS floats (~33 KB)

    dim3 grid(NTCH, NYB, NXB);
    gs_residual_partial<<<grid, THREADS, 0, stream>>>(src, ws);
    gs_reduce_mean<<<2, THREADS, 0, stream>>>(ws, out);
}